// LinearInternalBody_16844861735566
// MI455X (gfx1250) — compile-verified
//
#include <hip/hip_runtime.h>

// ---------------------------------------------------------------------------
// z[b,s,c,o] = sum_i c_out[b,s,c,i] * Wc[c,o,i]
//   with y = x @ Wn[c]^T (per-channel 2x2), amp = y0^2+y1^2, c_out = amp*y.
//
// View x as [N=131072 rows, 128 cols] (row = (b,s), col = 2c+d).
// Channel groups of 8 -> 16-col slabs; per-channel 2x2 weights packed into a
// 16x16 block-diagonal matrix; first linear done as 4x V_WMMA_F32_16X16X4_F32.
// Nonlinearity + second linear done in WMMA D-layout; the re/im partner lives
// in the adjacent lane -> one DPP quad_perm swap per value (pure VALU).
// x tile staged via async global->LDS copies (ASYNCcnt), LDS rows padded to
// 132 dwords so strided A-operand reads are bank-conflict free.
// ---------------------------------------------------------------------------

typedef __attribute__((ext_vector_type(2))) float v2f;
typedef __attribute__((ext_vector_type(4))) float v4f;
typedef __attribute__((ext_vector_type(8))) float v8f;
typedef __attribute__((ext_vector_type(4))) int   v4i;

typedef __attribute__((address_space(1))) v4i gas_v4i;  // global (__device__)
typedef __attribute__((address_space(3))) v4i las_v4i;  // LDS (__shared__)

#define ROW_DW       128   // floats per (b,s) row  (C*D = 64*2)
#define LDS_ROW_DW   132   // padded: 132 % 64 == 4 -> conflict-free strided reads
#define TILE_ROWS    64    // rows per block
#define BLOCK_THREADS 256  // 8 waves; wave w owns channel-group g = w

// Swap adjacent lanes (0<->1, 2<->3, ...) via DPP quad_perm:[1,0,3,2].
__device__ __forceinline__ float swap_pair(float v) {
    int s = __builtin_amdgcn_mov_dpp(__float_as_int(v),
                                     /*dpp_ctrl=*/0xB1,   // quad_perm 1,0,3,2
                                     /*row_mask=*/0xF, /*bank_mask=*/0xF,
                                     /*bound_ctrl=*/true);
    return __int_as_float(s);
}

__global__ __launch_bounds__(BLOCK_THREADS)
void quad_nonlin_kernel(const float* __restrict__ x,
                        const float* __restrict__ Wn,   // [C][2][2]
                        const float* __restrict__ Wc,   // [C][2][2]
                        float* __restrict__ out)
{
    __shared__ float tile[TILE_ROWS * LDS_ROW_DW];

    const int tid  = threadIdx.x;
    const long long row0 = (long long)blockIdx.x * TILE_ROWS;
    const float* gsrc = x + row0 * ROW_DW;

    // ---- Stage x tile (64 rows x first 128B) into LDS, coalesced 16B/lane --------
    // 64 rows * 8 b128-chunks = 512 chunks; 2 per thread.
    #pragma unroll
    for (int c = tid; c < TILE_ROWS * 8; c += BLOCK_THREADS) {
        const int r = c >> 3;
        const int s = c & 7;
        const float* gp = gsrc + r * ROW_DW + s * 4;
        float*       lp = &tile[r * LDS_ROW_DW + s * 4];
#if __has_builtin(__builtin_amdgcn_global_load_async_to_lds_b128)
        __builtin_amdgcn_global_load_async_to_lds_b128(
            (gas_v4i*)gp, (las_v4i*)lp, /*offset=*/0, /*cpol=*/0);
#else
        *(v4f*)lp = *(const v4f*)gp;
#endif
    }
#if __has_builtin(__builtin_amdgcn_global_load_async_to_lds_b128)
#if __has_builtin(__builtin_amdgcn_s_wait_asynccnt)
    __builtin_amdgcn_s_wait_asynccnt(0);
#else
    asm volatile("s_wait_asynccnt 0" ::: "memory");
#endif
#endif
    __syncthreads();

    // ---- Per-lane constants -----------------------------------------------
    const int lane = tid & 31;
    const int g    = tid >> 5;        // channel group (0..7), one per wave
    const int n    = lane & 15;       // column within group (output col N)
    const int half = lane >> 4;       // 0: lanes 0-15, 1: lanes 16-31
    const int ch   = g * 8 + (n >> 1);// absolute channel
    const int o    = n & 1;           // component (re/im)

    // Block-diagonal B operands for the first linear (V_WMMA B-layout:
    // lanes 0-15 hold K-rows {0,1} in vgpr {0,1}; lanes 16-31 hold rows {2,3}).
    // Wb[2a+i][2a+o] = Wn[a][o][i]; B_k covers absolute K-rows 4k..4k+3.
    v2f wpair;
    wpair.x = Wn[ch * 4 + o * 2 + 0];   // i = 0
    wpair.y = Wn[ch * 4 + o * 2 + 1];   // i = 1
    const v2f zero2 = {0.0f, 0.0f};
    v2f Bm[4];
    #pragma unroll
    for (int k = 0; k < 4; ++k) {
        const bool cond = ((n >> 1) == (2 * k + half));  // same 2x2 block?
        Bm[k] = cond ? wpair : zero2;
    }

    // Second linear collapses to two per-lane scalars (pair lives in lanes
    // n and n^1): z = amp * (y*wc_self + y_pair*wc_cross)
    const float wc_self  = Wc[ch * 4 + o * 2 + o];
    const float wc_cross = Wc[ch * 4 + o * 2 + (1 - o)];

    // ---- 4 sub-tiles of 16 rows each --------------------------------------
    #pragma unroll
    for (int m0 = 0; m0 < TILE_ROWS; m0 += 16) {
        // A-layout (16x4 f32): lane = M + 16*(K>=2), vgpr = K%2.
        // Lane reads float2 at cols (4k + 2*half) of its row.
        const float* lbase =
            &tile[(m0 + n) * LDS_ROW_DW + g * 16 + half * 2];
        v2f a0 = *(const v2f*)(lbase + 0);
        v2f a1 = *(const v2f*)(lbase + 4);
        v2f a2 = *(const v2f*)(lbase + 8);
        v2f a3 = *(const v2f*)(lbase + 12);

        v8f acc = {0.f, 0.f, 0.f, 0.f, 0.f, 0.f, 0.f, 0.f};
        acc = __builtin_amdgcn_wmma_f32_16x16x4_f32(false, a0, false, Bm[0],
                                                    (short)0, acc, false, false);
        acc = __builtin_amdgcn_wmma_f32_16x16x4_f32(false, a1, false, Bm[1],
                                                    (short)0, acc, false, false);
        acc = __builtin_amdgcn_wmma_f32_16x16x4_f32(false, a2, false, Bm[2],
                                                    (short)0, acc, false, false);
        acc = __builtin_amdgcn_wmma_f32_16x16x4_f32(false, a3, false, Bm[3],
                                                    (short)0, acc, false, false);

        // D-layout: lane holds col n; vgpr v holds row m = v + 8*half.
        // amp/nonlinearity/second linear elementwise; partner via DPP swap.
        const long long orow = row0 + m0;
        #pragma unroll
        for (int v = 0; v < 8; ++v) {
            const float y  = acc[v];
            const float ys = swap_pair(y);               // partner component
            const float amp = y * y + ys * ys;
            const float z   = amp * (y * wc_self + ys * wc_cross);
            const long long r = orow + v + 8 * half;
            out[r * ROW_DW + g * 16 + n] = z;
        }
    }
}

extern "C" void kernel_launch(void* const* d_in, const int* in_sizes, int n_in,
                              void* d_out, int out_size, void* d_ws, size_t ws_size,
                              hipStream_t stream)
{
    const float* x  = (const float*)d_in[0];   // [B,S,C,D] fp32
    const float* Wn = (const float*)d_in[1];   // [C,D,D]   fp32
    const float* Wc = (const float*)d_in[2];   // [C,D,D]   fp32
    float* out = (float*)d_out;

    const long long n_rows = (long long)in_sizes[0] / ROW_DW;  // B*S = 131072
    const int blocks = (int)(n_rows / TILE_ROWS);              // 2048

    quad_nonlin_kernel<<<dim3(blocks), dim3(BLOCK_THREADS), 0, stream>>>(
        x, Wn, Wc, out);
}